// RegionAggregator_66571993088895
// MI455X (gfx1250) — compile-verified
//
#include <hip/hip_runtime.h>
#include <hip/hip_bf16.h>

// CDNA5 / gfx1250. wave32. WMMA f32 16x16x4 fragments:
//   A (16x4, 2 VGPRs): lane L -> M = L%16, kh = L/16; a = {A[M][kb+2kh], A[M][kb+2kh+1]}
//   B (4x16, 2 VGPRs): lane L -> N = L%16, kh = L/16; b = {B[kb+2kh][N], B[kb+2kh+1][N]}
//   C/D (16x16, 8 VGPRs): VGPR g -> row = g + 8*(L/16), col = L%16
typedef __attribute__((ext_vector_type(2))) float v2f;
typedef __attribute__((ext_vector_type(8))) float v8f;

#define LDS_STRIDE 68   // 64 + 4 pad: 16B-aligned rows, banks = (4*row + col) % 64
#define SCALE 0.35355339059327373f  // (64/8)^-0.5

// ---------------------------------------------------------------------------
// Kernel 1: fold the four 64x64 weights into two:
//   W1[d][n]  = sum_e Wq[e][d] * Wk[e][n]      (qm = Q @ W1)
//   W2[d][n]  = sum_e Wv[e][d] * Wo[n][e]      (out = ctx @ W2 + bo)
// Tiny (2 * 64*64*64 MACs) — one block, plain VALU.
// ---------------------------------------------------------------------------
__global__ __launch_bounds__(256)
void prep_weights(const float* __restrict__ Wq, const float* __restrict__ Wk,
                  const float* __restrict__ Wv, const float* __restrict__ Wo,
                  float* __restrict__ W1, float* __restrict__ W2) {
    for (int idx = threadIdx.x; idx < 64 * 64; idx += 256) {
        const int d = idx >> 6;
        const int n = idx & 63;
        float s1 = 0.f, s2 = 0.f;
#pragma unroll 8
        for (int e = 0; e < 64; ++e) {
            s1 += Wq[e * 64 + d] * Wk[e * 64 + n];
            s2 += Wv[e * 64 + d] * Wo[n * 64 + e];
        }
        W1[idx] = s1;
        W2[idx] = s2;
    }
}

// ---------------------------------------------------------------------------
// Kernel 2: fused attention. One workgroup = 16 tokens, 256 threads = 8 waves.
//   Stage A: qm_tile(16x64)  = Q_tile @ W1           (WMMA f32 16x16x4, waves 0-3)
//   Stage B: per token: scores = R@qm, softmax(32), ctx = attn@R   (streams 256MB)
//   Stage C: out_tile(16x64) = ctx_tile @ W2 + bo    (WMMA f32 16x16x4, waves 0-3)
// HBM-bound: ~268 MB total traffic -> ~11.5 us floor at 23.3 TB/s.
// ---------------------------------------------------------------------------
__global__ __launch_bounds__(256)
void region_agg_main(const float* __restrict__ queries,
                     const float* __restrict__ regions,
                     const float* __restrict__ W1,
                     const float* __restrict__ W2,
                     const float* __restrict__ bo,
                     float* __restrict__ out) {
    __shared__ float lds_q[16 * LDS_STRIDE];
    __shared__ float lds_qm[16 * LDS_STRIDE];
    __shared__ float lds_ctx[16 * LDS_STRIDE];
    __shared__ float lds_attn[8 * 32];
    __shared__ float lds_R[8][32 * LDS_STRIDE];   // one 32x64 R buffer per wave

    const int tid  = threadIdx.x;
    const int lane = tid & 31;
    const int wave = tid >> 5;
    const int t0   = blockIdx.x * 16;             // first token of this WG

    // ---- cooperative load of Q tile (16x64 = 1024 floats, one float4/thread)
    {
        const int e   = tid * 4;
        const int row = e >> 6;
        const int col = e & 63;
        float4 qv = *(const float4*)(queries + (size_t)t0 * 64 + e);
        *(float4*)&lds_q[row * LDS_STRIDE + col] = qv;
    }
    __syncthreads();

    const int nn = lane & 15;    // M (A-frag) / N (B-frag) index
    const int kh = lane >> 4;    // K-half select

    // ---- Stage A: qm = Q @ W1 ; wave n owns columns [16n, 16n+16)
    if (wave < 4) {
        v8f c = {};
#pragma unroll
        for (int ks = 0; ks < 16; ++ks) {
            const int kr = 4 * ks + 2 * kh;
            v2f a, b;
            a.x = lds_q[nn * LDS_STRIDE + kr];
            a.y = lds_q[nn * LDS_STRIDE + kr + 1];
            b.x = W1[kr * 64 + wave * 16 + nn];
            b.y = W1[(kr + 1) * 64 + wave * 16 + nn];
            c = __builtin_amdgcn_wmma_f32_16x16x4_f32(
                    false, a, false, b, (short)0, c, false, false);
        }
#pragma unroll
        for (int g = 0; g < 8; ++g) {
            const int row = g + 8 * kh;
            lds_qm[row * LDS_STRIDE + wave * 16 + nn] = c[g];
        }
    }
    __syncthreads();

    // ---- Stage B: each wave handles 2 tokens sequentially
    float* myR = &lds_R[wave][0];
#pragma unroll 1
    for (int sub = 0; sub < 2; ++sub) {
        const int m = wave * 2 + sub;                 // token row in tile
        const size_t t = (size_t)(t0 + m);
        const float* Rg = regions + t * 2048;         // contiguous 8KB per token

        // coalesced global b128 -> LDS (stride-68 rows, 16B aligned)
#pragma unroll
        for (int j = 0; j < 16; ++j) {
            const int e   = (j * 32 + lane) * 4;
            float4 rv = *(const float4*)(Rg + e);
            const int row = e >> 6;
            const int col = e & 63;
            *(float4*)&myR[row * LDS_STRIDE + col] = rv;
        }
        __builtin_amdgcn_wave_barrier();

        // phase 1: lane r -> score_r = R[r,:] . qm[m,:]   (qm read is broadcast)
        float acc = 0.f;
        const float* rrow  = &myR[lane * LDS_STRIDE];
        const float* qmrow = &lds_qm[m * LDS_STRIDE];
#pragma unroll
        for (int j = 0; j < 16; ++j) {
            float4 r4 = *(const float4*)(rrow + 4 * j);
            float4 q4 = *(const float4*)(qmrow + 4 * j);
            acc += r4.x * q4.x + r4.y * q4.y + r4.z * q4.z + r4.w * q4.w;
        }
        const float score = acc * SCALE;

        // softmax across the 32 lanes of this wave
        float mx = score;
#pragma unroll
        for (int off = 16; off > 0; off >>= 1)
            mx = fmaxf(mx, __shfl_xor(mx, off, 32));
        const float ex = __expf(score - mx);
        float sm = ex;
#pragma unroll
        for (int off = 16; off > 0; off >>= 1)
            sm += __shfl_xor(sm, off, 32);
        const float attn = ex / sm;

        lds_attn[wave * 32 + lane] = attn;
        __builtin_amdgcn_wave_barrier();   // same-wave DS ops are in-order

        // phase 2: ctx[m, j] = sum_r attn_r * R[r, j]  (lane j owns cols j, j+32)
        float c0 = 0.f, c1 = 0.f;
#pragma unroll
        for (int r = 0; r < 32; ++r) {
            const float a = lds_attn[wave * 32 + r];          // broadcast
            c0 += a * myR[r * LDS_STRIDE + lane];             // conflict-free
            c1 += a * myR[r * LDS_STRIDE + lane + 32];
        }
        lds_ctx[m * LDS_STRIDE + lane]      = c0;
        lds_ctx[m * LDS_STRIDE + lane + 32] = c1;
    }
    __syncthreads();

    // ---- Stage C: out = ctx @ W2 + bo ; wave n owns columns [16n, 16n+16)
    if (wave < 4) {
        v8f c = {};
#pragma unroll
        for (int ks = 0; ks < 16; ++ks) {
            const int kr = 4 * ks + 2 * kh;
            v2f a, b;
            a.x = lds_ctx[nn * LDS_STRIDE + kr];
            a.y = lds_ctx[nn * LDS_STRIDE + kr + 1];
            b.x = W2[kr * 64 + wave * 16 + nn];
            b.y = W2[(kr + 1) * 64 + wave * 16 + nn];
            c = __builtin_amdgcn_wmma_f32_16x16x4_f32(
                    false, a, false, b, (short)0, c, false, false);
        }
        const float bias = bo[wave * 16 + nn];
#pragma unroll
        for (int g = 0; g < 8; ++g) {
            const int row = g + 8 * kh;
            out[((size_t)(t0 + row)) * 64 + wave * 16 + nn] = c[g] + bias;
        }
    }
}

// ---------------------------------------------------------------------------
extern "C" void kernel_launch(void* const* d_in, const int* in_sizes, int n_in,
                              void* d_out, int out_size, void* d_ws, size_t ws_size,
                              hipStream_t stream) {
    const float* queries = (const float*)d_in[0];  // [b,h,l,64]
    const float* regions = (const float*)d_in[1];  // [b,h,l,32,64]
    const float* Wq      = (const float*)d_in[2];
    const float* Wk      = (const float*)d_in[3];
    const float* Wv      = (const float*)d_in[4];
    const float* Wo      = (const float*)d_in[5];
    const float* bo      = (const float*)d_in[6];
    float* out = (float*)d_out;

    float* W1 = (float*)d_ws;          // 64x64
    float* W2 = W1 + 64 * 64;          // 64x64

    const int tokens = in_sizes[0] / 64;   // b*h*l = 32768

    prep_weights<<<1, 256, 0, stream>>>(Wq, Wk, Wv, Wo, W1, W2);
    region_agg_main<<<tokens / 16, 256, 0, stream>>>(queries, regions, W1, W2, bo, out);
}